// LogcumsumexpOp_73710228734210
// MI455X (gfx1250) — compile-verified
//
#include <hip/hip_runtime.h>
#include <math.h>

// Logcumsumexp along axis 1 of x:(8, 4096, 1024) f32.
//
// Roofline: 268 MB min HBM traffic -> ~11.5 us @ 23.3 TB/s. The naive
// per-element logaddexp puts VALU work right at that floor, so this version
// minimizes VALU/transcendentals per element:
//   Pass 1: chunk max (VALU-only loop) + sum of exp2(fma(...)) (1 trans/elem)
//           + one log per chunk. Second read of the chunk hits L0/L2.
//   Pass 2: normalized running sum: 2 trans + 3 VALU per element, carry-in
//           folded in via exp2(carry - m) with exp2(-inf)=0 as identity.
// Input (134 MB) stays resident in the 192 MB L2 between passes; output is
// written with non-temporal stores so it does not evict the input.

typedef float v4f __attribute__((ext_vector_type(4)));

#define BB      8
#define TT      4096
#define CC      1024
#define C4      (CC / 4)      // 256 float4 columns -> blockDim (8 wave32s)
#define NCHUNK  64
#define CT      (TT / NCHUNK) // 64 timesteps per chunk
#define PF      8             // prefetch distance (rows ahead)

#define NEG_INF (-__builtin_inff())
#define L2E     1.4426950408889634f   // log2(e)
#define LN2     0.6931471805599453f

// Pairwise logaddexp (used only for the short carry reduction, off the
// critical path). exp(-inf)=0 makes -inf the identity element.
__device__ __forceinline__ float laddexp(float a, float b) {
    float m = fmaxf(a, b);
    float d = fabsf(a - b);
    return m + __logf(1.0f + __expf(-d));
}

__device__ __forceinline__ v4f laddexp4(v4f a, v4f b) {
    v4f r;
    r.x = laddexp(a.x, b.x);
    r.y = laddexp(a.y, b.y);
    r.z = laddexp(a.z, b.z);
    r.w = laddexp(a.w, b.w);
    return r;
}

__device__ __forceinline__ v4f max4(v4f a, v4f b) {
    v4f r;
    r.x = fmaxf(a.x, b.x);
    r.y = fmaxf(a.y, b.y);
    r.z = fmaxf(a.z, b.z);
    r.w = fmaxf(a.w, b.w);
    return r;
}

// p = exp(x - m) computed as exp2(x*log2e - ml2), ml2 = m*log2e. 1 FMA + 1 trans.
__device__ __forceinline__ v4f expm4(v4f x, v4f ml2) {
    v4f r;
    r.x = __builtin_exp2f(__builtin_fmaf(x.x, L2E, -ml2.x));
    r.y = __builtin_exp2f(__builtin_fmaf(x.y, L2E, -ml2.y));
    r.z = __builtin_exp2f(__builtin_fmaf(x.z, L2E, -ml2.z));
    r.w = __builtin_exp2f(__builtin_fmaf(x.w, L2E, -ml2.w));
    return r;
}

// m + log(s) computed as fma(log2(s), ln2, m). 1 trans + 1 FMA.
__device__ __forceinline__ v4f mlog4(v4f s, v4f m) {
    v4f r;
    r.x = __builtin_fmaf(__log2f(s.x), LN2, m.x);
    r.y = __builtin_fmaf(__log2f(s.y), LN2, m.y);
    r.z = __builtin_fmaf(__log2f(s.z), LN2, m.z);
    r.w = __builtin_fmaf(__log2f(s.w), LN2, m.w);
    return r;
}

// Pass 1: per-chunk max and logsumexp total. Block = one (batch, chunk) pair
// over all 1024 columns. First loop (HBM read, warms L2) is VALU-cheap max;
// second loop re-reads from cache and does 1 transcendental per element.
__global__ void __launch_bounds__(C4)
lse_chunk_totals(const float* __restrict__ x, float* __restrict__ S,
                 float* __restrict__ M) {
    const int c4 = threadIdx.x;            // 0..255  (16B column group)
    const int k  = blockIdx.x % NCHUNK;    // chunk index along scan axis
    const int b  = blockIdx.x / NCHUNK;    // batch

    const v4f* __restrict__ xp =
        (const v4f*)(x + (size_t)(b * TT + k * CT) * CC) + c4;

    // Loop A: chunk max (pulls chunk from HBM into L2/L0).
    v4f m = {NEG_INF, NEG_INF, NEG_INF, NEG_INF};
#pragma unroll 8
    for (int t = 0; t < CT; ++t) {
        if (t + PF < CT)
            __builtin_prefetch((const void*)(xp + (size_t)(t + PF) * C4), 0, 3);
        m = max4(m, xp[(size_t)t * C4]);   // global_load_b128, coalesced
    }

    // Loop B: sum exp(x - m); re-read hits L0/L2.
    v4f ml2 = m * L2E;
    v4f s = {0.f, 0.f, 0.f, 0.f};
#pragma unroll 8
    for (int t = 0; t < CT; ++t)
        s += expm4(xp[(size_t)t * C4], ml2);

    // One log per chunk. S/M layout: [b][k][c], 2 MB each -> L2-resident.
    const size_t row = (size_t)(b * NCHUNK + k) * CC;
    ((v4f*)(S + row))[c4] = mlog4(s, m);
    ((v4f*)(M + row))[c4] = m;
}

// Pass 2: build the exclusive carry from chunk totals (<=63 pairwise combines
// over an L2-resident buffer), then run the normalized cumulative sum over the
// chunk (input re-read hits L2) and stream results out with NT stores.
__global__ void __launch_bounds__(C4)
scan_chunk_with_carry(const float* __restrict__ x, const float* __restrict__ S,
                      const float* __restrict__ M, float* __restrict__ out) {
    const int c4 = threadIdx.x;
    const int k  = blockIdx.x % NCHUNK;
    const int b  = blockIdx.x / NCHUNK;

    // Exclusive prefix (logsumexp of all preceding chunks); -inf for k==0.
    const v4f* __restrict__ Sp = (const v4f*)(S + (size_t)b * NCHUNK * CC) + c4;
    v4f carry = {NEG_INF, NEG_INF, NEG_INF, NEG_INF};
    for (int kk = 0; kk < k; ++kk)
        carry = laddexp4(carry, Sp[(size_t)kk * C4]);

    // Normalizer: max of (chunk max, carry). exp2(-inf)=0 handles k==0.
    const size_t row = (size_t)(b * NCHUNK + k) * CC;
    v4f m   = max4(((const v4f*)(M + row))[c4], carry);
    v4f ml2 = m * L2E;
    v4f s   = expm4(carry, ml2);           // exp(carry - m) in [0,1]

    const v4f* __restrict__ xp =
        (const v4f*)(x + (size_t)(b * TT + k * CT) * CC) + c4;
    v4f* __restrict__ op =
        (v4f*)(out + (size_t)(b * TT + k * CT) * CC) + c4;

#pragma unroll 4
    for (int t = 0; t < CT; ++t) {
        if (t + PF < CT)
            __builtin_prefetch((const void*)(xp + (size_t)(t + PF) * C4), 0, 3);
        s += expm4(xp[(size_t)t * C4], ml2);            // L2 hit (pass-1 warm)
        // NT store: don't let the 134 MB output evict the input from L2.
        __builtin_nontemporal_store(mlog4(s, m), op + (size_t)t * C4);
    }
}

extern "C" void kernel_launch(void* const* d_in, const int* in_sizes, int n_in,
                              void* d_out, int out_size, void* d_ws, size_t ws_size,
                              hipStream_t stream) {
    (void)in_sizes; (void)n_in; (void)out_size; (void)ws_size;

    const float* x   = (const float*)d_in[0];
    float*       out = (float*)d_out;
    float*       S   = (float*)d_ws;                       // 2 MB chunk totals
    float*       M   = (float*)d_ws + (size_t)BB * NCHUNK * CC; // 2 MB chunk maxes

    const dim3 grid(BB * NCHUNK);      // 512 blocks
    const dim3 block(C4);              // 256 threads = 8 wave32s

    lse_chunk_totals<<<grid, block, 0, stream>>>(x, S, M);
    scan_chunk_with_carry<<<grid, block, 0, stream>>>(x, S, M, out);
}